// DepthDC3x3_3_37151467110635
// MI455X (gfx1250) — compile-verified
//
#include <hip/hip_runtime.h>

typedef __attribute__((ext_vector_type(2))) float v2f;
typedef __attribute__((ext_vector_type(8))) float v8f;
typedef int v4i_ __attribute__((vector_size(4 * sizeof(int))));

#define KN 8
#define KC 64
#define KH 128
#define KW 128

#if __has_builtin(__builtin_amdgcn_global_load_async_to_lds_b128) && \
    __has_builtin(__builtin_amdgcn_s_wait_asynccnt)
#define HAS_ASYNC 1
#else
#define HAS_ASYNC 0
#endif

#if HAS_ASYNC
__device__ __forceinline__ void g2l_b128(const float* g, float* l) {
  // per-lane: copies 16 bytes from global address g to LDS address l, tracked by ASYNCcnt
  __builtin_amdgcn_global_load_async_to_lds_b128(
      (__attribute__((address_space(1))) v4i_*)(g),
      (__attribute__((address_space(3))) v4i_*)(l), 0, 0);
}
__device__ __forceinline__ void async_wait0() {
  __builtin_amdgcn_s_wait_asynccnt(0);
}
#endif

__device__ __forceinline__ v8f wmma4(v2f a, v2f b, v8f c) {
  // D = A(16x4 f32) * B(4x16 f32) + C(16x16 f32)
  return __builtin_amdgcn_wmma_f32_16x16x4_f32(
      /*neg_a=*/false, a, /*neg_b=*/false, b,
      /*c_mod=*/(short)0, c, /*reuse_a=*/false, /*reuse_b=*/false);
}

// -------------------------------------------------------------------------
// One-shot repack: OIHW [64][64][3][3] -> [tap][oc][ic] so each per-tap
// 16KB weight slab is contiguous (enables coalesced b128 async LDS fills).
// -------------------------------------------------------------------------
__global__ __launch_bounds__(256) void repack_w(const float* __restrict__ w,
                                                float* __restrict__ wr) {
  int i = blockIdx.x * 256 + threadIdx.x;      // over 9*64*64 = 36864
  if (i < 9 * KC * KC) {
    int tap = i >> 12;                         // /4096
    int rem = i & 4095;
    int o = rem >> 6;
    int ic = rem & 63;
    wr[i] = w[o * (KC * 9) + ic * 9 + tap];
  }
}

// -------------------------------------------------------------------------
// 3x3 conv, pad=1, 64ch -> 64ch, NCHW fp32, repacked weights [tap][oc][ic].
// Block = 256 threads (8 waves), one (n,h) row, 32-pixel tile.
// Double-buffered async weight slab: fill tap+1 overlaps tap's 16 WMMAs.
// -------------------------------------------------------------------------
__global__ __launch_bounds__(256) void conv3x3_wmma(
    const float* __restrict__ in, const float* __restrict__ wr,
    const float* __restrict__ bias, float* __restrict__ out) {
  __shared__ float ysl[KC * 3 * 34];   // input slab w/ halo (26 KB)
  __shared__ float wsl[2][KC * KC];    // double-buffered tap weights (32 KB)

  const int tid  = threadIdx.x;
  const int lane = tid & 31;
  const int wid  = tid >> 5;

  int bid = blockIdx.x;
  const int wt = bid & 3;              // W/32 = 4 tiles per row
  bid >>= 2;
  const int h = bid & (KH - 1);
  const int n = bid >> 7;
  const int w0 = wt * 32;

  // Zero-padded input slab (manual: halo needs synthesized zeros).
  for (int e = tid; e < KC * 3 * 34; e += 256) {
    int c = e / 102;
    int rem = e - c * 102;
    int r = rem / 34;
    int col = rem - r * 34;
    int gh = h + r - 1;
    int gw = w0 + col - 1;
    float v = 0.0f;
    if ((unsigned)gh < KH && (unsigned)gw < KW)
      v = in[((n * KC + c) * KH + gh) * KW + gw];
    ysl[e] = v;
  }

  // Prologue: fill weight buffer 0 (tap 0).
#if HAS_ASYNC
  for (int e = tid * 4; e < KC * KC; e += 1024) g2l_b128(&wr[e], &wsl[0][e]);
  async_wait0();
#else
  for (int e = tid; e < KC * KC; e += 256) wsl[0][e] = wr[e];
#endif
  __syncthreads();

  const int oc_base = (wid >> 1) * 16;
  const int px_base = (wid & 1) * 16;
  const int mrow  = lane & 15;          // A: m row / B: n col
  const int khalf = (lane >> 4) * 2;    // K pair per lane-half
  const int mh    = lane >> 4;

  v8f acc = {};

  for (int tap = 0; tap < 9; ++tap) {
    const int cur = tap & 1;
#if HAS_ASYNC
    if (tap < 8) {                      // overlap next tap's fill with compute
      const float* wt_nxt = &wr[(tap + 1) * KC * KC];
      for (int e = tid * 4; e < KC * KC; e += 1024)
        g2l_b128(&wt_nxt[e], &wsl[cur ^ 1][e]);
    }
#endif
    const int ky = tap / 3;
    const int kx = tap - ky * 3;
    const float* brow = &ysl[ky * 34 + px_base + mrow + kx];
    const float* arow = &wsl[cur][(oc_base + mrow) * KC + khalf];
#pragma unroll
    for (int k0 = 0; k0 < KC; k0 += 4) {
      v2f a = *(const v2f*)&arow[k0];
      v2f b;
      b.x = brow[(k0 + khalf) * 102];
      b.y = brow[(k0 + khalf + 1) * 102];
      acc = wmma4(a, b, acc);
    }
#if HAS_ASYNC
    async_wait0();                      // own async fills complete
    __syncthreads();                    // all waves done reading cur + filled nxt
#else
    __syncthreads();
    if (tap < 8) {
      const float* wt_nxt = &wr[(tap + 1) * KC * KC];
      for (int e = tid; e < KC * KC; e += 256) wsl[cur ^ 1][e] = wt_nxt[e];
    }
    __syncthreads();
#endif
  }

  // Epilogue: D vgpr r holds M = r + 8*(lane>=16), N = lane&15.
#pragma unroll
  for (int r = 0; r < 8; ++r) {
    int oc = oc_base + r + 8 * mh;
    int w  = w0 + px_base + (lane & 15);
    out[((n * KC + oc) * KH + h) * KW + w] = acc[r] + bias[oc];
  }
}

// -------------------------------------------------------------------------
// Fused: per-pixel dynamic kernel (GEMM M=576,K=64,N=16px via WMMA) kept in
// LDS, immediately consumed by the dynamic depthwise 3x3 — the 302MB kernel
// tensor never touches HBM.  Block = 256 threads, 16-pixel row tile.
// -------------------------------------------------------------------------
__global__ __launch_bounds__(256) void dynk_fused(
    const float* __restrict__ x, const float* __restrict__ t,
    const float* __restrict__ w2, const float* __restrict__ b2,
    float* __restrict__ res) {
  __shared__ float tsl[KC * 16];      // t activations [c][p]       (4 KB)
  __shared__ float xsl[KC * 3 * 18];  // x slab w/ halo [c][3][18]  (13.5 KB)
  __shared__ float ksl[576 * 16];     // dynamic kernel [m][p]      (36 KB)

  const int tid  = threadIdx.x;
  const int lane = tid & 31;
  const int wid  = tid >> 5;

  int bid = blockIdx.x;
  const int wt = bid & 7;             // W/16 = 8 tiles per row
  bid >>= 3;
  const int h = bid & (KH - 1);
  const int n = bid >> 7;
  const int w0 = wt * 16;

  // t slab: always in-bounds, contiguous 16B per lane -> async b128.
#if HAS_ASYNC
  {
    int c = tid >> 2;
    int p4 = (tid & 3) * 4;
    g2l_b128(&t[((n * KC + c) * KH + h) * KW + w0 + p4], &tsl[c * 16 + p4]);
  }
#else
  for (int e = tid; e < KC * 16; e += 256) {
    int c = e >> 4;
    int p = e & 15;
    tsl[e] = t[((n * KC + c) * KH + h) * KW + w0 + p];
  }
#endif
  // x slab: halo needs zeros -> manual.
  for (int e = tid; e < KC * 3 * 18; e += 256) {
    int c = e / 54;
    int rem = e - c * 54;
    int r = rem / 18;
    int col = rem - r * 18;
    int gh = h + r - 1;
    int gw = w0 + col - 1;
    float v = 0.0f;
    if ((unsigned)gh < KH && (unsigned)gw < KW)
      v = x[((n * KC + c) * KH + gh) * KW + gw];
    xsl[e] = v;
  }
#if HAS_ASYNC
  async_wait0();
#endif
  __syncthreads();

  const int mrow  = lane & 15;
  const int khalf = (lane >> 4) * 2;
  const int mh    = lane >> 4;

  // 576/16 = 36 M-tiles distributed over 8 waves.
  for (int mt = wid; mt < 36; mt += 8) {
    v8f acc = {};
    const float* arow = &w2[(mt * 16 + mrow) * KC + khalf];  // L2-resident
    const float* bcol = &tsl[khalf * 16 + mrow];
#pragma unroll
    for (int k0 = 0; k0 < KC; k0 += 4) {
      v2f a = *(const v2f*)&arow[k0];
      v2f b;
      b.x = bcol[k0 * 16];
      b.y = bcol[(k0 + 1) * 16];
      acc = wmma4(a, b, acc);
    }
#pragma unroll
    for (int r = 0; r < 8; ++r) {
      int m = mt * 16 + r + 8 * mh;
      ksl[m * 16 + (lane & 15)] = acc[r] + b2[m];
    }
  }
  __syncthreads();

  // Dynamic depthwise 3x3: 64ch x 16px = 1024 outputs, 4 per thread.
  for (int i = 0; i < 4; ++i) {
    int idx = tid + i * 256;
    int c = idx >> 4;
    int p = idx & 15;
    float sum = 0.0f;
#pragma unroll
    for (int ky = 0; ky < 3; ++ky)
#pragma unroll
      for (int kx = 0; kx < 3; ++kx)
        sum += xsl[c * 54 + ky * 18 + p + kx] *
               ksl[(c * 9 + ky * 3 + kx) * 16 + p];
    res[((n * KC + c) * KH + h) * KW + w0 + p] = sum;
  }
}

extern "C" void kernel_launch(void* const* d_in, const int* in_sizes, int n_in,
                              void* d_out, int out_size, void* d_ws, size_t ws_size,
                              hipStream_t stream) {
  (void)in_sizes; (void)n_in; (void)out_size; (void)ws_size;
  const float* x      = (const float*)d_in[0];
  const float* y      = (const float*)d_in[1];
  const float* w_gk1  = (const float*)d_in[2];
  const float* b_gk1  = (const float*)d_in[3];
  const float* w_gk2  = (const float*)d_in[4];
  const float* b_gk2  = (const float*)d_in[5];
  const float* w_fuse = (const float*)d_in[6];
  const float* b_fuse = (const float*)d_in[7];
  float* out = (float*)d_out;

  const size_t nchw = (size_t)KN * KC * KH * KW;
  float* t    = (float*)d_ws;          // [8,64,128,128]
  float* res  = t + nchw;              // [8,64,128,128]
  float* wr1  = res + nchw;            // [9][64][64] repacked w_gk1
  float* wrF  = wr1 + 9 * KC * KC;     // [9][64][64] repacked w_fuse

  dim3 blk(256);
  repack_w<<<(9 * KC * KC + 255) / 256, blk, 0, stream>>>(w_gk1, wr1);
  repack_w<<<(9 * KC * KC + 255) / 256, blk, 0, stream>>>(w_fuse, wrF);
  // Stage A: t = conv3x3(y, w_gk1) + b_gk1
  conv3x3_wmma<<<KN * KH * (KW / 32), blk, 0, stream>>>(y, wr1, b_gk1, t);
  // Stage B+C fused: per-pixel dynamic kernel + dynamic depthwise on x
  dynk_fused<<<KN * KH * (KW / 16), blk, 0, stream>>>(x, t, w_gk2, b_gk2, res);
  // Stage D: out = conv3x3(res, w_fuse) + b_fuse
  conv3x3_wmma<<<KN * KH * (KW / 32), blk, 0, stream>>>(res, wrF, b_fuse, out);
}